// GraphSAGE_2164663517731
// MI455X (gfx1250) — compile-verified
//
#include <hip/hip_runtime.h>

#define D 128
#define LDSTRIDE 132  // 128 + 4 pad: avoids 64-bank conflicts on column reads

typedef float v2f __attribute__((ext_vector_type(2)));
typedef float v8f __attribute__((ext_vector_type(8)));

// ---------------------------------------------------------------- utilities

__global__ void zero_f_kernel(float* __restrict__ p, int n) {
    int i = blockIdx.x * blockDim.x + threadIdx.x;
    if (i < n) p[i] = 0.0f;
}

__global__ void degree_kernel(const int* __restrict__ dst, float* __restrict__ deg, int e) {
    int i = blockIdx.x * blockDim.x + threadIdx.x;
    if (i < e) atomicAdd(&deg[dst[i]], 1.0f);
}

__global__ void invdeg_kernel(float* __restrict__ deg, int n) {
    int i = blockIdx.x * blockDim.x + threadIdx.x;
    if (i < n) deg[i] = 1.0f / fmaxf(deg[i], 1.0f);
}

// ------------------------------------------------- edge scatter (mean-agg numerator)
// One wave per edge: 32 lanes x float4 = 128 contiguous floats. Coalesced
// 512B gather of x[src], 128 global_atomic_add_f32 into agg[dst].

__global__ __launch_bounds__(256) void scatter_kernel(
    const float* __restrict__ x, const int* __restrict__ src,
    const int* __restrict__ dst, float* __restrict__ agg, int e) {
    long long t = (long long)blockIdx.x * blockDim.x + threadIdx.x;
    int edge  = (int)(t >> 5);
    int chunk = (int)(t & 31);
    if (edge >= e) return;
    int s = src[edge];
    int d = dst[edge];
    const float4 v = *(const float4*)(x + (size_t)s * D + chunk * 4);
    float* p = agg + (size_t)d * D + chunk * 4;
    atomicAdd(p + 0, v.x);
    atomicAdd(p + 1, v.y);
    atomicAdd(p + 2, v.z);
    atomicAdd(p + 3, v.w);
}

// ------------------------------------------------- fused SAGE GEMM via WMMA
// out[i,:] = (agg[i,:]*invdeg[i]) @ Wl^T + h[i,:] @ Wr^T + b   (+ ReLU)
// Block = 16 nodes x 128 outputs. 8 waves, each owns a 16x16 output tile.
// fp32 matrix path: v_wmma_f32_16x16x4_f32, 64 WMMA per wave (2 x K=128).

__global__ __launch_bounds__(256) void sage_gemm_kernel(
    const float* __restrict__ hin, const float* __restrict__ agg,
    const float* __restrict__ invdeg, const float* __restrict__ Wl,
    const float* __restrict__ Wr, const float* __restrict__ bias,
    float* __restrict__ out, int n, int relu) {
    __shared__ float sm[16 * LDSTRIDE];  // mean-aggregated tile
    __shared__ float sh[16 * LDSTRIDE];  // self tile

    const int tid   = threadIdx.x;
    const int node0 = blockIdx.x * 16;

    // Stage 16x128 tiles into LDS; fold the 1/deg scaling into the load.
    for (int idx = tid; idx < 16 * D; idx += 256) {
        int r = idx >> 7;
        int c = idx & (D - 1);
        int node = node0 + r;
        float vm = 0.0f, vh = 0.0f;
        if (node < n) {
            float s = invdeg[node];
            vm = agg[(size_t)node * D + c] * s;
            vh = hin[(size_t)node * D + c];
        }
        sm[r * LDSTRIDE + c] = vm;
        sh[r * LDSTRIDE + c] = vh;
    }
    __syncthreads();

    const int wave  = tid >> 5;
    const int lane  = tid & 31;
    const int half  = lane >> 4;        // 0: lanes 0-15, 1: lanes 16-31
    const int arow  = lane & 15;        // A-matrix row / D-matrix column index
    const int kof   = half << 1;        // K sub-offset per ISA f32 A/B layout
    const int ocol  = wave * 16 + arow; // this lane's output column

    const float* smp = sm + arow * LDSTRIDE + kof;
    const float* shp = sh + arow * LDSTRIDE + kof;
    const float* wlp = Wl + (size_t)ocol * D + kof;  // B[k][o] = Wl[o*D + k]
    const float* wrp = Wr + (size_t)ocol * D + kof;

    v8f acc = {};

    // agg-tile x Wl^T
#pragma unroll
    for (int k = 0; k < D; k += 4) {
        v2f a, b;
        a[0] = smp[k];  a[1] = smp[k + 1];
        b[0] = wlp[k];  b[1] = wlp[k + 1];
        acc = __builtin_amdgcn_wmma_f32_16x16x4_f32(
            false, a, false, b, (short)0, acc, false, false);
    }
    // self-tile x Wr^T
#pragma unroll
    for (int k = 0; k < D; k += 4) {
        v2f a, b;
        a[0] = shp[k];  a[1] = shp[k + 1];
        b[0] = wrp[k];  b[1] = wrp[k + 1];
        acc = __builtin_amdgcn_wmma_f32_16x16x4_f32(
            false, a, false, b, (short)0, acc, false, false);
    }

    // 16x16 f32 C/D layout: VGPR r -> M=r (lanes 0-15) / M=r+8 (lanes 16-31),
    // N = ocol. Fuse bias + ReLU on the way out.
    const float bv    = bias[ocol];
    const int   mbase = half << 3;
#pragma unroll
    for (int r = 0; r < 8; ++r) {
        int node = node0 + mbase + r;
        if (node < n) {
            float v = acc[r] + bv;
            if (relu) v = fmaxf(v, 0.0f);
            out[(size_t)node * D + ocol] = v;
        }
    }
}

// ---------------------------------------------------------------- driver

extern "C" void kernel_launch(void* const* d_in, const int* in_sizes, int n_in,
                              void* d_out, int out_size, void* d_ws, size_t ws_size,
                              hipStream_t stream) {
    const float* x  = (const float*)d_in[0];
    const int*   ei = (const int*)d_in[1];
    const int n = in_sizes[0] / D;
    const int e = in_sizes[1] / 2;
    const int* src = ei;
    const int* dst = ei + e;

    const float* Wl[3] = {(const float*)d_in[2], (const float*)d_in[5], (const float*)d_in[8]};
    const float* Wr[3] = {(const float*)d_in[3], (const float*)d_in[6], (const float*)d_in[9]};
    const float* bb[3] = {(const float*)d_in[4], (const float*)d_in[7], (const float*)d_in[10]};

    const size_t nd_bytes = (size_t)n * D * sizeof(float);
    char*  ws     = (char*)d_ws;
    float* agg    = (float*)(ws);
    float* h1     = (float*)(ws + nd_bytes);
    float* h2     = (float*)(ws + 2 * nd_bytes);
    float* invdeg = (float*)(ws + 3 * nd_bytes);

    const int T = 256;
    const int nd        = n * D;
    const int gz_nd     = (nd + T - 1) / T;
    const int gz_n      = (n + T - 1) / T;
    const int gz_e      = (e + T - 1) / T;
    const int g_scatter = (int)(((long long)e * 32 + T - 1) / T);
    const int g_gemm    = (n + 15) / 16;

    // Degree -> 1/max(deg,1), computed once (edge structure is layer-invariant).
    zero_f_kernel<<<gz_n, T, 0, stream>>>(invdeg, n);
    degree_kernel<<<gz_e, T, 0, stream>>>(dst, invdeg, e);
    invdeg_kernel<<<gz_n, T, 0, stream>>>(invdeg, n);

    const float* cur = x;
    float* outs[3] = {h1, h2, (float*)d_out};
    for (int l = 0; l < 3; ++l) {
        zero_f_kernel<<<gz_nd, T, 0, stream>>>(agg, nd);
        scatter_kernel<<<g_scatter, T, 0, stream>>>(cur, src, dst, agg, e);
        sage_gemm_kernel<<<g_gemm, T, 0, stream>>>(
            cur, agg, invdeg, Wl[l], Wr[l], bb[l], outs[l], n, (l < 2) ? 1 : 0);
        cur = outs[l];
    }
}